// BiologicalBrain_25288767438884
// MI455X (gfx1250) — compile-verified
//
#include <hip/hip_runtime.h>

#define EPT 1024   // edges per tile (1024 edges * 4B = 4KB per array = 256 thr * 16B)
#define TPB 256

// Region offsets (from reference SIZES)
#define OFF_V1 0
#define OFF_V2 50000
#define OFF_A1 80000
#define OFF_T  100000
#define OFF_P  130000
#define OFF_PF 150000
#define OFF_H  180000
#define OFF_M  200000
#define TOTAL  210000

#define BITS_WORDS_PER_REGION 2048   // 8KB per region, >= ceil(50000/32)=1563
#define N_REGIONS 8
#define MAX_BIT_WORDS 1568           // LDS staging capacity (>= 1563)

// LIF constants (float32 of the reference's doubles)
#define LEAK_F        0.9512294245007140f   // exp(-1/20)
#define ADAPT_DECAY_F 0.9900498337491681f   // exp(-1/100)
#define RATE_DECAY_F  0.9990004998333750f   // exp(-1/1000)
#define RATE_GAIN_F   0.000999500166625f    // 1 - exp(-1/1000)

__global__ void zero_kernel(unsigned* __restrict__ p, int n) {
    int i = blockIdx.x * blockDim.x + threadIdx.x;
    if (i < n) p[i] = 0u;
}

// One LIF region update. Iext != nullptr -> external drive (region-local index),
// else I = 0.5 * acc[g] (OUT_SCALE folded here; multi-input regions sum into the
// same accumulator, which distributes over the 0.5).
// Spikes are emitted as a region-local bitmask (1 bit/neuron) for the scatters.
__global__ void lif_kernel(const float* __restrict__ Iext,
                           const float* __restrict__ acc,
                           const float* __restrict__ v_in,
                           const float* __restrict__ a_in,
                           const float* __restrict__ thr,
                           const float* __restrict__ r_in,
                           unsigned* __restrict__ bits,    // region-local bitmask
                           float* __restrict__ v_out,
                           float* __restrict__ a_out,
                           float* __restrict__ r_out,
                           float* __restrict__ spike_copy, // motor only
                           int n, int off) {
    int i = blockIdx.x * blockDim.x + threadIdx.x;
    if (i >= n) return;
    int g = off + i;
    float I = Iext ? Iext[i] : 0.5f * acc[g];
    float v = v_in[g] * LEAK_F + I;
    float a = a_in[g];
    float s = (v >= thr[g] + a) ? 1.0f : 0.0f;
    v = (s > 0.5f) ? 0.0f : v;
    a = a * ADAPT_DECAY_F + s * 0.1f;
    float r = r_in[g] * RATE_DECAY_F + s * RATE_GAIN_F;
    v_out[g] = v;
    a_out[g] = a;
    r_out[g] = r;
    if (spike_copy) spike_copy[i] = s;

#if __has_builtin(__builtin_amdgcn_ballot_w32)
    // wave32: one bitmask word per 32 neurons; i is region-local and waves are
    // 32-aligned within the region, so lane (i&31)==0 owns word i>>5.
    unsigned m = __builtin_amdgcn_ballot_w32(s > 0.5f);
    if ((i & 31) == 0) bits[i >> 5] = m;
#else
    if (s > 0.5f) atomicOr(&bits[i >> 5], 1u << (i & 31));  // bits pre-zeroed
#endif
}

// Sparse scatter: for each edge e with pre-neuron spike bit set, accumulate
// (e < n_inh ? -4*w[e] : w[e]) into acc[post[e]].
// Edge streams (pre, post, w) are staged tile-by-tile into LDS via the CDNA5
// async-LDS path (global_load_async_to_lds_b128, ASYNCcnt), double-buffered.
// Each thread stages & consumes exactly its own 16B chunk of each array, so
// the only sync needed for the tiles is s_wait_asynccnt. The pre-region spike
// bitmask (<= 6.3KB) is staged once into LDS, making the per-edge spike test
// an LDS read: all remaining global traffic is pure streaming + L2 atomics.
__global__ __launch_bounds__(TPB) void scatter_kernel(
    const int*      __restrict__ pre,
    const int*      __restrict__ post,
    const float*    __restrict__ w,
    const unsigned* __restrict__ bits,  // pre-region spike bitmask
    float*          __restrict__ acc,   // acc + post_region_offset
    int n, int n_inh, int nwords) {
    __shared__ int      s_pre [2][EPT];
    __shared__ int      s_post[2][EPT];
    __shared__ float    s_w   [2][EPT];
    __shared__ unsigned s_bit [MAX_BIT_WORDS];

    const int tid = threadIdx.x;
    const int c4  = tid * 4;         // this thread's 4-edge slot in the tile
    const int nt  = n / EPT;         // number of full tiles

    // LDS byte addresses (flat-pointer low 32 bits == LDS offset)
    unsigned lp0 = (unsigned)(unsigned long long)&s_pre [0][c4];
    unsigned lp1 = (unsigned)(unsigned long long)&s_pre [1][c4];
    unsigned lq0 = (unsigned)(unsigned long long)&s_post[0][c4];
    unsigned lq1 = (unsigned)(unsigned long long)&s_post[1][c4];
    unsigned lw0 = (unsigned)(unsigned long long)&s_w   [0][c4];
    unsigned lw1 = (unsigned)(unsigned long long)&s_w   [1][c4];

    auto issue_tile = [&](int buf, int tile) {
        unsigned long long gp = (unsigned long long)(pre  + tile * EPT + c4);
        unsigned long long gq = (unsigned long long)(post + tile * EPT + c4);
        unsigned long long gw = (unsigned long long)(w    + tile * EPT + c4);
        asm volatile("global_load_async_to_lds_b128 %0, %1, off"
                     :: "v"(buf ? lp1 : lp0), "v"(gp) : "memory");
        asm volatile("global_load_async_to_lds_b128 %0, %1, off"
                     :: "v"(buf ? lq1 : lq0), "v"(gq) : "memory");
        asm volatile("global_load_async_to_lds_b128 %0, %1, off"
                     :: "v"(buf ? lw1 : lw0), "v"(gw) : "memory");
    };

    int t = blockIdx.x;              // uniform control flow -> full EXEC
    if (t < nt) issue_tile(0, t);    // overlap first tile with bitmask staging

    for (int j = tid; j < nwords; j += TPB) s_bit[j] = bits[j];
    __syncthreads();

    int buf = 0;
    for (; t < nt; t += gridDim.x) {
        int tn = t + (int)gridDim.x;
        if (tn < nt) {
            issue_tile(buf ^ 1, tn);
            // 6 async loads outstanding; first 3 (current tile) complete in order
            asm volatile("s_wait_asynccnt 0x3" ::: "memory");
        } else {
            asm volatile("s_wait_asynccnt 0x0" ::: "memory");
        }
        int base = t * EPT + c4;
#pragma unroll
        for (int k = 0; k < 4; ++k) {
            int p = s_pre[buf][c4 + k];
            if ((s_bit[p >> 5] >> (p & 31)) & 1u) {
                float wv = s_w[buf][c4 + k];
                float we = ((base + k) < n_inh) ? (-4.0f * wv) : wv;
                atomicAdd(&acc[s_post[buf][c4 + k]], we);
            }
        }
        buf ^= 1;
    }

    // Tail edges (n % EPT): direct bounds-checked loads (spike test still LDS)
    for (long long i = (long long)nt * EPT + (long long)blockIdx.x * TPB + tid;
         i < (long long)n;
         i += (long long)gridDim.x * TPB) {
        int p = pre[i];
        if ((s_bit[p >> 5] >> (p & 31)) & 1u) {
            float wv = w[i];
            float we = (i < (long long)n_inh) ? (-4.0f * wv) : wv;
            atomicAdd(&acc[post[i]], we);
        }
    }
}

extern "C" void kernel_launch(void* const* d_in, const int* in_sizes, int n_in,
                              void* d_out, int out_size, void* d_ws, size_t ws_size,
                              hipStream_t stream) {
    (void)n_in; (void)out_size; (void)ws_size;

    const float* visual = (const float*)d_in[0];
    const float* audio  = (const float*)d_in[1];
    const float* v_in   = (const float*)d_in[2];
    const float* a_in   = (const float*)d_in[3];
    const float* thr    = (const float*)d_in[4];
    const float* r_in   = (const float*)d_in[5];

    float* out_s = (float*)d_out;            // motor spikes (10000)
    float* out_v = out_s + 10000;            // v_new  (210000)
    float* out_a = out_v + TOTAL;            // a_new  (210000)
    float* out_r = out_a + TOTAL;            // r_new  (210000)

    float*    acc  = (float*)d_ws;                     // 210000 f32 currents
    unsigned* bits = (unsigned*)(acc + TOTAL);         // 8 * 2048 words

    // Zero accumulator + all region bitmasks in one pass
    const int zero_words = TOTAL + N_REGIONS * BITS_WORDS_PER_REGION;
    zero_kernel<<<(zero_words + TPB - 1) / TPB, TPB, 0, stream>>>(
        (unsigned*)d_ws, zero_words);

    // Region index for bitmask bases: v1,v2,a1,t,p,pf,h,m
    auto rbits = [&](int ridx) { return bits + ridx * BITS_WORDS_PER_REGION; };

    auto lif = [&](const float* Iext, int off, int n, int ridx, float* scopy) {
        lif_kernel<<<(n + TPB - 1) / TPB, TPB, 0, stream>>>(
            Iext, acc, v_in, a_in, thr, r_in,
            rbits(ridx), out_v, out_a, out_r, scopy, n, off);
    };
    auto scat = [&](int ci, int pre_ridx, int pre_size, int post_off) {
        const int*   pre  = (const int*)d_in[6 + 4 * ci];
        const int*   post = (const int*)d_in[7 + 4 * ci];
        const float* wgt  = (const float*)d_in[8 + 4 * ci];
        int n = in_sizes[6 + 4 * ci];
        int n_inh = (int)((double)n * 0.2);   // inh = first 20% of edges
        int nwords = (pre_size + 31) / 32;
        int blocks = (n + EPT - 1) / EPT;
        if (blocks > 2048) blocks = 2048;
        if (blocks < 1) blocks = 1;
        scatter_kernel<<<blocks, TPB, 0, stream>>>(
            pre, post, wgt, rbits(pre_ridx), acc + post_off, n, n_inh, nwords);
    };

    // Dependency order; stream ordering provides synchronization.
    lif(visual, OFF_V1, 50000, 0, nullptr);            // v1
    lif(audio,  OFF_A1, 20000, 2, nullptr);            // a1
    scat(0, 0, 50000, OFF_V2);                         // v1 -> v2
    lif(nullptr, OFF_V2, 30000, 1, nullptr);           // v2
    scat(1, 1, 30000, OFF_T);                          // v2 -> temporal
    scat(3, 2, 20000, OFF_T);                          // a1 -> temporal
    lif(nullptr, OFF_T, 30000, 3, nullptr);            // temporal
    scat(2, 1, 30000, OFF_P);                          // v2 -> parietal
    lif(nullptr, OFF_P, 20000, 4, nullptr);            // parietal
    scat(4, 3, 30000, OFF_PF);                         // temporal -> prefrontal
    scat(5, 4, 20000, OFF_PF);                         // parietal -> prefrontal
    lif(nullptr, OFF_PF, 30000, 5, nullptr);           // prefrontal
    scat(6, 3, 30000, OFF_H);                          // temporal -> hippocampus
    scat(7, 4, 20000, OFF_H);                          // parietal -> hippocampus
    lif(nullptr, OFF_H, 20000, 6, nullptr);            // hippocampus
    scat(8, 5, 30000, OFF_M);                          // prefrontal -> motor
    lif(nullptr, OFF_M, 10000, 7, out_s);              // motor (spikes -> d_out)
}